// ContextualAttention_CotLayer_attention_30382598652159
// MI455X (gfx1250) — compile-verified
//
#include <hip/hip_runtime.h>
#include <math.h>

typedef float v2f __attribute__((ext_vector_type(2)));
typedef float v8f __attribute__((ext_vector_type(8)));
typedef unsigned int u32x4 __attribute__((ext_vector_type(4)));
typedef int i32x8 __attribute__((ext_vector_type(8)));
typedef int i32x4 __attribute__((ext_vector_type(4)));

#define NPIX 16384
#define HDIM 128
#define WDIM 128
#define PXB  256          // pixels per block
#define BROW 257          // LDS row stride for B panel (256 + 1 TDM pad dword)

// ---------------------------------------------------------------------------
// Issue one TDM transfer: 16 rows x 256 px fp32 tile (row stride NPIX) from
// global into LDS at ldsByte, with 1-dword LDS padding after every 256 dwords
// (rows land at stride 257 floats -> bank-conflict-free reads).
// Descriptor per CDNA5 ISA (D# groups 0/1; 2D tile).
// ---------------------------------------------------------------------------
__device__ __forceinline__ void tdm_load_panel(const float* gsrc, unsigned ldsByte)
{
    unsigned long long ga = (unsigned long long)(size_t)gsrc;
    u32x4 g0;
    g0.x = 1u;                                        // count=1, user desc
    g0.y = ldsByte;                                   // lds_addr
    g0.z = (unsigned)ga;                              // global_addr[31:0]
    g0.w = (unsigned)((ga >> 32) & 0x1FFFFFFu)        // global_addr[56:32]
         | (2u << 30);                                // type=2 (image)
    i32x8 g1;
    g1[0] = (2 << 16)        // data_size = 4 bytes
          | (1 << 20)        // pad_enable
          | (7 << 22);       // pad_interval: 256 dwords ; pad_amount=0 -> 1 dword
    g1[1] = 0x4000 << 16;    // tensor_dim0 = 16384 (bits 63:48)
    g1[2] = 16 << 16;        // tensor_dim1 = 16     (bits 95:80)
    g1[3] = 256 << 16;       // tile_dim0  = 256     (bits 127:112)
    g1[4] = 16;              // tile_dim1  = 16      (bits 143:128)
    g1[5] = 16384;           // tensor_dim0_stride   (bits 191:160)
    g1[6] = 0;
    g1[7] = 0;
    i32x4 z4 = {0, 0, 0, 0};          // groups 2/3 unused (2D tile)
    i32x8 z8 = {0, 0, 0, 0, 0, 0, 0, 0};
    __builtin_amdgcn_tensor_load_to_lds(g0, g1, z4, z4, z8, 0);
}

// ---------------------------------------------------------------------------
// 1x1 conv as GEMM:  Y[co, p] = sum_ci W[co,ci] * X[ci,p] (+bias) (+lrelu)
// Block = 256 thr = 8 waves -> one 16(co) x 256(px) tile per block.
// A (16 x Ci weight tile) staged in LDS once, shared by all 8 waves.
// B panels (16 x 256) staged by the Tensor Data Mover, double-buffered,
// synchronized with s_wait_tensorcnt + workgroup barrier.
// Each wave: 16x32 output tile = 2 accumulators, 2 WMMA per K-step.
// Weights must be zero-padded so CoPad = 16*coTiles (see pad_w kernel).
// ---------------------------------------------------------------------------
__global__ __launch_bounds__(256) void gemm1x1_wmma(
    const float* __restrict__ X, const float* __restrict__ Wp,
    const float* __restrict__ bias, float* __restrict__ Y,
    int Ci, int Co, int coTiles,
    int inBatchStride, int outBatchStride, int outChOff, int act)
{
    __shared__ float sW[16 * 129];          // 16 x (Ci+1), Ci <= 128
    __shared__ float sB[2][16 * BROW];      // double-buffered TDM panels

    const int lane = threadIdx.x & 31;
    const int wave = threadIdx.x >> 5;
    const int half = lane >> 4;             // K half select
    const int l    = lane & 15;

    const int pxGroups = NPIX / PXB;        // 64 per batch
    int unit = blockIdx.x;                  // (b, coTile, pxGroup)
    int b    = unit / (coTiles * pxGroups);
    int rem  = unit % (coTiles * pxGroups);
    int coT  = rem / pxGroups;
    int px0  = (rem % pxGroups) * PXB;
    int co0  = coT * 16;

    const float* Xb = X + (size_t)b * inBatchStride + px0;
    const int CiP = Ci + 1;

    // cooperative A-tile load into LDS (no guards: weights are padded)
    for (int idx = threadIdx.x; idx < 16 * Ci; idx += 256) {
        int row = idx / Ci, col = idx % Ci;
        sW[row * CiP + col] = Wp[(size_t)(co0 + row) * Ci + col];
    }

    const int nPanels = Ci >> 4;
    if (wave == 0)
        tdm_load_panel(Xb, (unsigned)(size_t)(const void*)&sB[0][0]);

    v8f acc0 = {0.f,0.f,0.f,0.f,0.f,0.f,0.f,0.f};
    v8f acc1 = {0.f,0.f,0.f,0.f,0.f,0.f,0.f,0.f};
    const int wl = wave * 32 + l;           // pixel within block tile

    for (int p = 0; p < nPanels; ++p) {
        if (wave == 0)
            __builtin_amdgcn_s_wait_tensorcnt(0);   // panel p landed
        __syncthreads();                            // visible to all waves
        if (wave == 0 && p + 1 < nPanels)           // prefetch next panel
            tdm_load_panel(Xb + (size_t)((p + 1) << 4) * NPIX,
                           (unsigned)(size_t)(const void*)&sB[(p + 1) & 1][0]);

        const float* Bp = &sB[p & 1][0];
        const int kb = p << 4;
#pragma unroll
        for (int kk = 0; kk < 4; ++kk) {
            int kcol = kb + kk * 4 + 2 * half;
            v2f a;
            a.x = sW[l * CiP + kcol];
            a.y = sW[l * CiP + kcol + 1];
            int row = kk * 4 + 2 * half;
            v2f b0, b1;
            b0.x = Bp[row * BROW + wl];
            b0.y = Bp[(row + 1) * BROW + wl];
            b1.x = Bp[row * BROW + wl + 16];
            b1.y = Bp[(row + 1) * BROW + wl + 16];
            acc0 = __builtin_amdgcn_wmma_f32_16x16x4_f32(
                       false, a, false, b0, (short)0, acc0, false, false);
            acc1 = __builtin_amdgcn_wmma_f32_16x16x4_f32(
                       false, a, false, b1, (short)0, acc1, false, false);
        }
        __syncthreads();                            // done with buf[p&1]
    }

    float* Yb = Y + (size_t)b * outBatchStride + (size_t)outChOff * NPIX + px0;
#pragma unroll
    for (int r = 0; r < 8; ++r) {
        int co = co0 + r + 8 * half;
        if (co < Co) {
            float v0 = acc0[r], v1 = acc1[r];
            if (bias) { float bb = bias[co]; v0 += bb; v1 += bb; }
            if (act) {
                v0 = (v0 > 0.f) ? v0 : 0.1f * v0;
                v1 = (v1 > 0.f) ? v1 : 0.1f * v1;
            }
            Yb[(size_t)co * NPIX + wave * 32 + l]      = v0;
            Yb[(size_t)co * NPIX + wave * 32 + l + 16] = v1;
        }
    }
}

// ---------------------------------------------------------------------------
// Zero-pad a [Co x Ci] weight matrix to [CoPad x Ci] (row-major).
// ---------------------------------------------------------------------------
__global__ __launch_bounds__(256) void pad_w(
    const float* __restrict__ src, float* __restrict__ dst,
    int Co, int CoPad, int Ci)
{
    int i = blockIdx.x * blockDim.x + threadIdx.x;
    if (i >= CoPad * Ci) return;
    int r = i / Ci;
    dst[i] = (r < Co) ? src[i] : 0.f;
}

// ---------------------------------------------------------------------------
// Grouped 3x3 conv (groups=4, 16ch/group, pad=1) + leaky ReLU, dual output.
// One wave -> 16(co) x 16(px) tile; per tap a 16x16x16 GEMM via 4 f32 WMMAs.
// ---------------------------------------------------------------------------
__global__ __launch_bounds__(256) void gconv3x3_wmma(
    const float* __restrict__ X, const float* __restrict__ Wk,
    float* __restrict__ Y1, float* __restrict__ Y2)
{
    const int lane = threadIdx.x & 31;
    const int wave = threadIdx.x >> 5;
    const int half = lane >> 4;
    const int l    = lane & 15;

    int unit = blockIdx.x * 8 + wave;       // (b, group, pixel tile)
    int b  = unit / (4 * 1024);
    int r0 = unit % (4 * 1024);
    int g  = r0 / 1024;
    int pt = r0 % 1024;
    int p0 = pt * 16;
    int y  = p0 / WDIM;
    int x0 = p0 % WDIM;

    const float* Xb = X + (size_t)b * 64 * NPIX;

    v8f acc = {0.f,0.f,0.f,0.f,0.f,0.f,0.f,0.f};
    for (int dy = 0; dy < 3; ++dy) {
        int yy = y + dy - 1;
        bool yok = (yy >= 0) && (yy < HDIM);
        for (int dx = 0; dx < 3; ++dx) {
            int xx = x0 + l + dx - 1;
            bool ok = yok && (xx >= 0) && (xx < WDIM);
#pragma unroll
            for (int kk = 0; kk < 4; ++kk) {
                int ci = kk * 4 + 2 * half;
                int co = g * 16 + l;
                v2f a;
                a.x = Wk[(((size_t)co * 16 + ci    ) * 3 + dy) * 3 + dx];
                a.y = Wk[(((size_t)co * 16 + ci + 1) * 3 + dy) * 3 + dx];
                v2f bm;
                if (ok) {
                    const float* xp = Xb + (size_t)(g * 16 + ci) * NPIX + yy * WDIM + xx;
                    bm.x = xp[0];
                    bm.y = xp[NPIX];
                } else {
                    bm.x = 0.f; bm.y = 0.f;
                }
                acc = __builtin_amdgcn_wmma_f32_16x16x4_f32(
                          false, a, false, bm, (short)0, acc, false, false);
            }
        }
    }

    size_t obase = (size_t)b * 128 * NPIX + (size_t)64 * NPIX + p0 + l;
#pragma unroll
    for (int r = 0; r < 8; ++r) {
        int co = g * 16 + r + 8 * half;
        float v = acc[r];
        v = (v > 0.f) ? v : 0.1f * v;       // leaky relu 0.1
        Y1[obase + (size_t)co * NPIX] = v;
        Y2[obase + (size_t)co * NPIX] = v;
    }
}

// ---------------------------------------------------------------------------
// Per-pixel dynamic 3x3 local conv (SHARE=8). Optional residual add.
// ---------------------------------------------------------------------------
__global__ __launch_bounds__(256) void local_conv_k(
    const float* __restrict__ Hh, const float* __restrict__ Wd,
    const float* __restrict__ resid, float* __restrict__ Y,
    int outBatchStride, int outChOff)
{
    size_t idx = (size_t)blockIdx.x * blockDim.x + threadIdx.x;  // B*64*NPIX
    int x = idx & (WDIM - 1);
    int y = (idx >> 7) & (HDIM - 1);
    int c = (idx >> 14) & 63;
    int b = (int)(idx >> 20);

    const float* Hb = Hh + ((size_t)b * 64 + c) * NPIX;
    const float* Wb = Wd + ((size_t)b * 72 + (c >> 3) * 9) * NPIX + y * WDIM + x;

    float s = 0.f;
#pragma unroll
    for (int dy = 0; dy < 3; ++dy) {
        int yy = y + dy - 1;
#pragma unroll
        for (int dx = 0; dx < 3; ++dx) {
            int xx = x + dx - 1;
            float p = 0.f;
            if (yy >= 0 && yy < HDIM && xx >= 0 && xx < WDIM)
                p = Hb[yy * WDIM + xx];
            s += p * Wb[(size_t)(dy * 3 + dx) * NPIX];
        }
    }
    if (resid) s += resid[((size_t)b * 64 + c) * NPIX + y * WDIM + x];
    Y[(size_t)b * outBatchStride + (size_t)(outChOff + c) * NPIX + y * WDIM + x] = s;
}

// ---------------------------------------------------------------------------
// Softmax over the H axis of [B,72,H,W] (axis=2, as in the reference).
// ---------------------------------------------------------------------------
__global__ __launch_bounds__(256) void softmax_h(float* __restrict__ A, int total)
{
    int t = blockIdx.x * blockDim.x + threadIdx.x;
    if (t >= total) return;
    int x  = t & (WDIM - 1);
    int cb = t >> 7;                         // (b*72 + ch)
    float* col = A + (size_t)cb * NPIX + x;

    float m = -3.402823466e+38f;
    for (int y2 = 0; y2 < HDIM; ++y2) {
        float v = col[y2 * WDIM];
        m = (v > m) ? v : m;
    }
    float s = 0.f;
    for (int y2 = 0; y2 < HDIM; ++y2)
        s += __expf(col[y2 * WDIM] - m);
    float inv = 1.f / s;
    for (int y2 = 0; y2 < HDIM; ++y2)
        col[y2 * WDIM] = __expf(col[y2 * WDIM] - m) * inv;
}

// ---------------------------------------------------------------------------
extern "C" void kernel_launch(void* const* d_in, const int* in_sizes, int n_in,
                              void* d_out, int out_size, void* d_ws, size_t ws_size,
                              hipStream_t stream) {
    const float* x       = (const float*)d_in[0];
    const float* phi_w   = (const float*)d_in[1];
    const float* phi_b   = (const float*)d_in[2];
    const float* theta_w = (const float*)d_in[3];
    const float* theta_b = (const float*)d_in[4];
    const float* ke_w    = (const float*)d_in[5];
    const float* e1_w    = (const float*)d_in[6];
    const float* e2_w    = (const float*)d_in[7];
    const float* e2_b    = (const float*)d_in[8];
    const float* c1_w    = (const float*)d_in[9];
    const float* sw1_w   = (const float*)d_in[10];
    const float* sw2_w   = (const float*)d_in[11];
    const float* sw2_b   = (const float*)d_in[12];
    const float* sec_w   = (const float*)d_in[13];
    const float* sec_b   = (const float*)d_in[14];
    float* out = (float*)d_out;
    float* ws  = (float*)d_ws;

    const int B = 8;
    size_t off = 0;
    float* key0 = ws + off; off += (size_t)B * 64  * NPIX;  // phi output
    float* qk   = ws + off; off += (size_t)B * 128 * NPIX;  // [query | key]
    float* t32  = ws + off; off += (size_t)B * 32  * NPIX;  // embed hidden
    float* wdyn = ws + off; off += (size_t)B * 72  * NPIX;  // dynamic kernels 1
    float* c1b  = ws + off; off += (size_t)B * 64  * NPIX;  // conv1 output
    float* hcat = ws + off; off += (size_t)B * 128 * NPIX;  // [local | key]
    float* attn = ws + off; off += (size_t)B * 72  * NPIX;  // dynamic kernels 2
    float* secb = ws + off; off += (size_t)B * 64  * NPIX;  // sec conv output
    float* e2p  = ws + off; off += (size_t)80 * 32;         // padded e2_w
    float* sw2p = ws + off; off += (size_t)80 * 32;         // padded sw2_w

    dim3 blk(256);
    const int PG = NPIX / PXB;   // 64 pixel groups per batch

    // pad the Co=72 heads to Co=80 so GEMM A-loads are unconditional
    pad_w<<<(80 * 32 + 255) / 256, blk, 0, stream>>>(e2_w,  e2p,  72, 80, 32);
    pad_w<<<(80 * 32 + 255) / 256, blk, 0, stream>>>(sw2_w, sw2p, 72, 80, 32);

    // key0 = phi(x);  query -> qk[0:64]
    gemm1x1_wmma<<<B * 4 * PG, blk, 0, stream>>>(x, phi_w,   phi_b,   key0, 64, 64, 4,  64 * NPIX,  64 * NPIX, 0, 0);
    gemm1x1_wmma<<<B * 4 * PG, blk, 0, stream>>>(x, theta_w, theta_b, qk,   64, 64, 4,  64 * NPIX, 128 * NPIX, 0, 0);
    // key = lrelu(gconv3x3(key0)) -> qk[64:128] and hcat[64:128]
    gconv3x3_wmma<<<4096, blk, 0, stream>>>(key0, ke_w, qk, hcat);
    // dynamic kernels: w = e2(lrelu(e1(qk)))
    gemm1x1_wmma<<<B * 2 * PG, blk, 0, stream>>>(qk,  e1_w, nullptr, t32, 128, 32, 2, 128 * NPIX, 32 * NPIX, 0, 1);
    gemm1x1_wmma<<<B * 5 * PG, blk, 0, stream>>>(t32, e2p,  e2_b,   wdyn, 32, 72, 5,  32 * NPIX, 72 * NPIX, 0, 0);
    // h = local_conv(c1(x), w) -> hcat[0:64]
    gemm1x1_wmma<<<B * 4 * PG, blk, 0, stream>>>(x, c1_w, nullptr, c1b, 64, 64, 4, 64 * NPIX, 64 * NPIX, 0, 0);
    local_conv_k<<<32768, blk, 0, stream>>>(c1b, wdyn, nullptr, hcat, 128 * NPIX, 0);
    // second weights: attn = softmax_H(sw2(lrelu(sw1(hcat))))
    gemm1x1_wmma<<<B * 2 * PG, blk, 0, stream>>>(hcat, sw1_w, nullptr, t32, 128, 32, 2, 128 * NPIX, 32 * NPIX, 0, 1);
    gemm1x1_wmma<<<B * 5 * PG, blk, 0, stream>>>(t32,  sw2p, sw2_b,  attn, 32, 72, 5,  32 * NPIX, 72 * NPIX, 0, 0);
    softmax_h<<<(B * 72 * WDIM + 255) / 256, blk, 0, stream>>>(attn, B * 72 * WDIM);
    // out = local_conv(sec(hcat), attn) + x
    gemm1x1_wmma<<<B * 4 * PG, blk, 0, stream>>>(hcat, sec_w, sec_b, secb, 128, 64, 4, 128 * NPIX, 64 * NPIX, 0, 0);
    local_conv_k<<<32768, blk, 0, stream>>>(secb, attn, x, out, 64 * NPIX, 0);
}